// OctreePositionEmbedding_69604239999073
// MI455X (gfx1250) — compile-verified
//
#include <hip/hip_runtime.h>
#include <hip/hip_bf16.h>
#include <stdint.h>

#define NF 256
#define LN_EPS 1e-5f

// Native clang vector: legal for __builtin_nontemporal_store and gives
// elementwise ops + scalar broadcast. Lowers to b128 loads/stores.
typedef float vf4 __attribute__((ext_vector_type(4)));

__device__ __forceinline__ vf4 ld4(const float* p) {
    return *(const vf4*)p;
}

// One wave (32 lanes) handles one output row of 256 floats (8 f32 per lane,
// as two vf4's: columns [lane*4, lane*4+4) and [128+lane*4, ...)).
// Block = 256 threads = 8 waves = 8 rows.
template <int W, int Z>
__global__ __launch_bounds__(256)
void octree_pos_ln_kernel(const int* __restrict__ idx,
                          const float* __restrict__ rowE,
                          const float* __restrict__ colE,
                          const float* __restrict__ zeeE,
                          const float* __restrict__ gamma,
                          const float* __restrict__ beta,
                          float* __restrict__ out,
                          int n) {
    __shared__ float sG[NF];
    __shared__ float sB[NF];

    const int tid = threadIdx.x;

    // ---- CDNA5 async global->LDS staging of gamma/beta (ASYNCcnt path) ----
    // Wave 0 stages gamma (1 KB), wave 1 stages beta (1 KB); each lane issues
    // two b128 async transfers (per-lane LDS dest addr + 64-bit global addr).
    if (tid < 32) {
        uint32_t l0 = (uint32_t)(uintptr_t)(const void*)&sG[tid * 4];
        const float* g0 = gamma + tid * 4;
        asm volatile("global_load_async_to_lds_b128 %0, %1, off"
                     :: "v"(l0), "v"(g0) : "memory");
        uint32_t l1 = l0 + 512;
        const float* g1 = g0 + 128;
        asm volatile("global_load_async_to_lds_b128 %0, %1, off"
                     :: "v"(l1), "v"(g1) : "memory");
    } else if (tid < 64) {
        const int t = tid - 32;
        uint32_t l0 = (uint32_t)(uintptr_t)(const void*)&sB[t * 4];
        const float* b0 = beta + t * 4;
        asm volatile("global_load_async_to_lds_b128 %0, %1, off"
                     :: "v"(l0), "v"(b0) : "memory");
        uint32_t l1 = l0 + 512;
        const float* b1 = b0 + 128;
        asm volatile("global_load_async_to_lds_b128 %0, %1, off"
                     :: "v"(l1), "v"(b1) : "memory");
    }
    asm volatile("s_wait_asynccnt 0" ::: "memory");
    __syncthreads();

    const int wave = tid >> 5;
    const int lane = tid & 31;
    const int row  = blockIdx.x * 8 + wave;
    if (row >= n) return;

    // Decode flat index with compile-time W,Z (magic-multiply, no v_div).
    const int id  = idx[row];
    const int r   = id / (W * Z);
    const int rem = id - r * (W * Z);
    const int c   = rem / Z;
    const int k   = rem - c * Z;

    const int off0 = lane * 4;
    const int off1 = off0 + 128;

    const float* rp = rowE + r * NF;
    const float* cp = colE + c * NF;
    const float* zp = zeeE + k * NF;

    // Gathers: tables are tiny (<=426 KB/level) -> L2-resident on a 192MB L2.
    vf4 v0 = ld4(rp + off0) + ld4(cp + off0) + ld4(zp + off0);
    vf4 v1 = ld4(rp + off1) + ld4(cp + off1) + ld4(zp + off1);

    // Per-lane partial sum / sum-of-squares over the 8 owned elements.
    float s  = v0.x + v0.y + v0.z + v0.w + v1.x + v1.y + v1.z + v1.w;
    float sq = v0.x * v0.x + v0.y * v0.y + v0.z * v0.z + v0.w * v0.w +
               v1.x * v1.x + v1.y * v1.y + v1.z * v1.z + v1.w * v1.w;

    // wave32 butterfly reduction across the row's 32 lanes.
    #pragma unroll
    for (int m = 16; m >= 1; m >>= 1) {
        s  += __shfl_xor(s,  m, 32);
        sq += __shfl_xor(sq, m, 32);
    }

    const float mean = s * (1.0f / NF);
    const float var  = sq * (1.0f / NF) - mean * mean;
    const float rstd = __frsqrt_rn(var + LN_EPS);

    const vf4 g0 = ld4(&sG[off0]);
    const vf4 g1 = ld4(&sG[off1]);
    const vf4 b0 = ld4(&sB[off0]);
    const vf4 b1 = ld4(&sB[off1]);

    const vf4 o0 = (v0 - mean) * rstd * g0 + b0;
    const vf4 o1 = (v1 - mean) * rstd * g1 + b1;

    // Streaming output (561 MB total, written once) -> non-temporal b128 stores.
    float* op = out + (size_t)row * NF;
    __builtin_nontemporal_store(o0, (vf4*)(op + off0));
    __builtin_nontemporal_store(o1, (vf4*)(op + off1));
}

extern "C" void kernel_launch(void* const* d_in, const int* in_sizes, int n_in,
                              void* d_out, int out_size, void* d_ws, size_t ws_size,
                              hipStream_t stream) {
    (void)n_in; (void)out_size; (void)d_ws; (void)ws_size;

    const int* idx1 = (const int*)d_in[0];
    const int* idx2 = (const int*)d_in[1];
    const int* idx3 = (const int*)d_in[2];
    const float* row1 = (const float*)d_in[3];
    const float* col1 = (const float*)d_in[4];
    const float* zee1 = (const float*)d_in[5];
    const float* row2 = (const float*)d_in[6];
    const float* col2 = (const float*)d_in[7];
    const float* zee2 = (const float*)d_in[8];
    const float* row3 = (const float*)d_in[9];
    const float* col3 = (const float*)d_in[10];
    const float* zee3 = (const float*)d_in[11];
    const float* gamma = (const float*)d_in[12];
    const float* beta  = (const float*)d_in[13];

    float* out = (float*)d_out;

    const int n1 = in_sizes[0];   // 8000
    const int n2 = in_sizes[1];   // 60000
    const int n3 = in_sizes[2];   // 480000

    dim3 blk(256);

    // Level 1: (h,w,z) = (50,50,4)
    octree_pos_ln_kernel<50, 4><<<dim3((n1 + 7) / 8), blk, 0, stream>>>(
        idx1, row1, col1, zee1, gamma, beta, out, n1);
    // Level 2: (100,100,8)
    octree_pos_ln_kernel<100, 8><<<dim3((n2 + 7) / 8), blk, 0, stream>>>(
        idx2, row2, col2, zee2, gamma, beta, out + (size_t)n1 * NF, n2);
    // Level 3: (200,200,16)
    octree_pos_ln_kernel<200, 16><<<dim3((n3 + 7) / 8), blk, 0, stream>>>(
        idx3, row3, col3, zee3, gamma, beta, out + (size_t)(n1 + n2) * NF, n3);
}